// SAGE_46772193853511
// MI455X (gfx1250) — compile-verified
//
#include <hip/hip_runtime.h>
#include <math.h>

#define NNODES 100000
#define NEDGES 1600000
#define FIN 128
#define HID 64
#define NCLS 40

typedef __attribute__((ext_vector_type(2))) float v2f;
typedef __attribute__((ext_vector_type(8))) float v8f;

__global__ void zero_f32(float* __restrict__ p, int n) {
  int i = blockIdx.x * blockDim.x + threadIdx.x;
  if (i < n) p[i] = 0.0f;
}

__global__ void degree_kernel(const int* __restrict__ dst, float* __restrict__ deg) {
  int e = blockIdx.x * blockDim.x + threadIdx.x;
  if (e < NEDGES) atomicAdd(&deg[dst[e]], 1.0f);
}

// ---------------------------------------------------------------------------
// GEMM1: P = x @ W1_l, Q = x @ W1_r     x:[N,128] row-major, W:[128,64]
// One wave per 16-row tile; 8 column tiles of 16 (4 for P, 4 for Q);
// K loop of 32 steps of V_WMMA_F32_16X16X4_F32.
// A layout: lanes 0-15 hold M=0..15 K={0,1}; lanes 16-31 hold K={2,3}.
// C/D layout: element(M = v + 8*(lane/16), N = lane%16) in VGPR v.
// ---------------------------------------------------------------------------
__global__ void __launch_bounds__(256) gemm1_kernel(
    const float* __restrict__ x, const float* __restrict__ Wl,
    const float* __restrict__ Wr, float* __restrict__ P, float* __restrict__ Q) {
  const int lane = threadIdx.x & 31;
  const int wave = threadIdx.x >> 5;
  const int tile = blockIdx.x * (blockDim.x >> 5) + wave;
  if (tile >= NNODES / 16) return;           // wave-uniform: EXEC stays all-1s
  const int row0 = tile * 16;
  const int half = lane >> 4;
  const int lrow = lane & 15;

  const float* xrow = x + (size_t)(row0 + lrow) * FIN + 2 * half;
  v2f a[FIN / 4];
#pragma unroll
  for (int k = 0; k < FIN / 4; ++k) {
    a[k].x = xrow[4 * k + 0];
    a[k].y = xrow[4 * k + 1];
  }

#pragma unroll
  for (int ct = 0; ct < 8; ++ct) {
    const float* W = (ct < 4) ? Wl : Wr;
    float* O = (ct < 4) ? P : Q;
    const int n0 = (ct & 3) * 16;
    const float* wp = W + (size_t)(2 * half) * HID + n0 + lrow;
    v8f c = {};
#pragma unroll
    for (int k = 0; k < FIN / 4; ++k) {
      v2f b;
      b.x = wp[(size_t)(4 * k) * HID];
      b.y = wp[(size_t)(4 * k + 1) * HID];
      c = __builtin_amdgcn_wmma_f32_16x16x4_f32(false, a[k], false, b,
                                                (short)0, c, false, false);
    }
#pragma unroll
    for (int v = 0; v < 8; ++v)
      O[(size_t)(row0 + v + 8 * half) * HID + n0 + lrow] = c[v];
  }
}

// ---------------------------------------------------------------------------
// GEMM2: P = h @ W2_l, Q = h @ W2_r     h:[N,64], W:[64,40]; 40 padded to 48.
// ---------------------------------------------------------------------------
__global__ void __launch_bounds__(256) gemm2_kernel(
    const float* __restrict__ h, const float* __restrict__ Wl,
    const float* __restrict__ Wr, float* __restrict__ P, float* __restrict__ Q) {
  const int lane = threadIdx.x & 31;
  const int wave = threadIdx.x >> 5;
  const int tile = blockIdx.x * (blockDim.x >> 5) + wave;
  if (tile >= NNODES / 16) return;
  const int row0 = tile * 16;
  const int half = lane >> 4;
  const int lrow = lane & 15;

  const float* hrow = h + (size_t)(row0 + lrow) * HID + 2 * half;
  v2f a[HID / 4];
#pragma unroll
  for (int k = 0; k < HID / 4; ++k) {
    a[k].x = hrow[4 * k + 0];
    a[k].y = hrow[4 * k + 1];
  }

#pragma unroll
  for (int ct = 0; ct < 6; ++ct) {
    const float* W = (ct < 3) ? Wl : Wr;
    float* O = (ct < 3) ? P : Q;
    const int n0 = (ct % 3) * 16;
    const int n = n0 + lrow;
    v8f c = {};
#pragma unroll
    for (int k = 0; k < HID / 4; ++k) {
      const int krow = 4 * k + 2 * half;
      v2f b;
      b.x = (n < NCLS) ? W[(size_t)krow * NCLS + n] : 0.0f;
      b.y = (n < NCLS) ? W[(size_t)(krow + 1) * NCLS + n] : 0.0f;
      c = __builtin_amdgcn_wmma_f32_16x16x4_f32(false, a[k], false, b,
                                                (short)0, c, false, false);
    }
    if (n < NCLS) {
#pragma unroll
      for (int v = 0; v < 8; ++v)
        O[(size_t)(row0 + v + 8 * half) * NCLS + n] = c[v];
    }
  }
}

// agg[dst] += P[src], 64 floats/edge, 16 threads/edge * float4
__global__ void scatter1_kernel(const int* __restrict__ src, const int* __restrict__ dst,
                                const float* __restrict__ P, float* __restrict__ agg) {
  int g = blockIdx.x * blockDim.x + threadIdx.x;
  if (g >= NEDGES * 16) return;
  int e = g >> 4;
  int c = g & 15;
  int s = src[e], d = dst[e];
  const float4 v = *reinterpret_cast<const float4*>(P + (size_t)s * HID + c * 4);
  float* o = agg + (size_t)d * HID + c * 4;
  atomicAdd(o + 0, v.x);
  atomicAdd(o + 1, v.y);
  atomicAdd(o + 2, v.z);
  atomicAdd(o + 3, v.w);
}

// agg[dst] += P[src], 40 floats/edge, 10 threads/edge * float4
__global__ void scatter2_kernel(const int* __restrict__ src, const int* __restrict__ dst,
                                const float* __restrict__ P, float* __restrict__ agg) {
  int g = blockIdx.x * blockDim.x + threadIdx.x;
  if (g >= NEDGES * 10) return;
  int e = g / 10;
  int c = g - e * 10;
  int s = src[e], d = dst[e];
  const float4 v = *reinterpret_cast<const float4*>(P + (size_t)s * NCLS + c * 4);
  float* o = agg + (size_t)d * NCLS + c * 4;
  atomicAdd(o + 0, v.x);
  atomicAdd(o + 1, v.y);
  atomicAdd(o + 2, v.z);
  atomicAdd(o + 3, v.w);
}

// h = relu(agg/deg + Q1 + b1), written in place over Q1
__global__ void relu_combine_kernel(const float* __restrict__ agg,
                                    const float* __restrict__ deg,
                                    const float* __restrict__ b1,
                                    float* __restrict__ hq) {
  int g = blockIdx.x * blockDim.x + threadIdx.x;
  if (g >= NNODES * HID) return;
  int i = g >> 6;          // / HID
  int j = g & (HID - 1);
  float inv = 1.0f / fmaxf(deg[i], 1.0f);
  float v = agg[g] * inv + hq[g] + b1[j];
  hq[g] = fmaxf(v, 0.0f);
}

// out holds agg2 sums; finish: v = agg2/deg + Q2 + b2 then log_softmax.
// One wave per node; wave32 shuffle reductions.
__global__ void __launch_bounds__(256) logsoftmax_kernel(
    const float* __restrict__ deg, const float* __restrict__ Q2,
    const float* __restrict__ b2, float* __restrict__ out) {
  const int lane = threadIdx.x & 31;
  const int wave = threadIdx.x >> 5;
  const int i = blockIdx.x * (blockDim.x >> 5) + wave;
  if (i >= NNODES) return;
  const float inv = 1.0f / fmaxf(deg[i], 1.0f);
  const size_t base = (size_t)i * NCLS;

  float v0 = out[base + lane] * inv + Q2[base + lane] + b2[lane];
  float v1 = -INFINITY;
  if (lane < NCLS - 32)
    v1 = out[base + lane + 32] * inv + Q2[base + lane + 32] + b2[lane + 32];

  float m = fmaxf(v0, v1);
#pragma unroll
  for (int off = 16; off >= 1; off >>= 1) m = fmaxf(m, __shfl_xor(m, off, 32));
  float s = __expf(v0 - m) + ((lane < NCLS - 32) ? __expf(v1 - m) : 0.0f);
#pragma unroll
  for (int off = 16; off >= 1; off >>= 1) s += __shfl_xor(s, off, 32);
  const float L = m + __logf(s);

  out[base + lane] = v0 - L;
  if (lane < NCLS - 32) out[base + lane + 32] = v1 - L;
}

extern "C" void kernel_launch(void* const* d_in, const int* in_sizes, int n_in,
                              void* d_out, int out_size, void* d_ws, size_t ws_size,
                              hipStream_t stream) {
  const float* x   = (const float*)d_in[0];
  const int*   ei  = (const int*)d_in[1];
  const float* W1l = (const float*)d_in[2];
  const float* W1r = (const float*)d_in[3];
  const float* b1  = (const float*)d_in[4];
  const float* W2l = (const float*)d_in[5];
  const float* W2r = (const float*)d_in[6];
  const float* b2  = (const float*)d_in[7];
  float* out = (float*)d_out;
  const int* src = ei;
  const int* dst = ei + NEDGES;

  // workspace layout (floats): deg[N] | P[N*64] (P1->P2) | Hq[N*64] (Q1->h) |
  //                            AgQ[N*64] (agg1->Q2)   total = 193*N floats
  float* ws  = (float*)d_ws;
  float* deg = ws;
  float* P   = ws + NNODES;
  float* Hq  = P + (size_t)NNODES * HID;
  float* AgQ = Hq + (size_t)NNODES * HID;

  const int B = 256;
  zero_f32<<<(NNODES + B - 1) / B, B, 0, stream>>>(deg, NNODES);
  zero_f32<<<(NNODES * HID + B - 1) / B, B, 0, stream>>>(AgQ, NNODES * HID);
  zero_f32<<<(NNODES * NCLS + B - 1) / B, B, 0, stream>>>(out, NNODES * NCLS);

  degree_kernel<<<(NEDGES + B - 1) / B, B, 0, stream>>>(dst, deg);

  gemm1_kernel<<<(NNODES / 16 + 7) / 8, B, 0, stream>>>(x, W1l, W1r, P, Hq);
  scatter1_kernel<<<(NEDGES * 16 + B - 1) / B, B, 0, stream>>>(src, dst, P, AgQ);
  relu_combine_kernel<<<(NNODES * HID + B - 1) / B, B, 0, stream>>>(AgQ, deg, b1, Hq);

  gemm2_kernel<<<(NNODES / 16 + 7) / 8, B, 0, stream>>>(Hq, W2l, W2r, P, AgQ);
  scatter2_kernel<<<(NEDGES * 10 + B - 1) / B, B, 0, stream>>>(src, dst, P, out);

  logsoftmax_kernel<<<(NNODES + 7) / 8, B, 0, stream>>>(deg, AgQ, b2, out);
}